// BidirectionalAdditiveAttention_52158082842796
// MI455X (gfx1250) — compile-verified
//
#include <hip/hip_runtime.h>
#include <hip/hip_bf16.h>
#include <math.h>

typedef __attribute__((ext_vector_type(2))) float v2f;
typedef __attribute__((ext_vector_type(8))) float v8f;
typedef int v4i __attribute__((ext_vector_type(4)));
typedef __attribute__((address_space(1))) v4i gv4i;   // global (AS1)
typedef __attribute__((address_space(3))) v4i lv4i;   // LDS (AS3)

#define B_  4
#define N1_ 256
#define N2_ 1024
#define K_  256
#define A_  128
#define V_  256

// ---------------------------------------------------------------------------
// Generic fp32 WMMA GEMM: C[b] = A[b] (M x K, row stride lda)
//                              * B[b] (element (k,n) at k*bs_k + n*bs_n)
//                              + bias[n] (optional)
// One wave computes a 64x16 C tile (4 stacked 16x16 accumulators) so the
// B fragment is loaded once per k-step and reused by 4 WMMAs.
// Requires M % 64 == 0 (true for all call sites).
// ---------------------------------------------------------------------------
__global__ __launch_bounds__(256)
void gemm_f32_wmma(const float* __restrict__ Am, const float* __restrict__ Bm,
                   const float* __restrict__ bias, float* __restrict__ Cm,
                   long a_batch, long b_batch, long c_batch,
                   int Kdim, int lda, int bs_k, int bs_n, int ldc,
                   int tiles_m, int tiles_n, int total_waves)
{
    int wave = (int)((blockIdx.x * blockDim.x + threadIdx.x) >> 5);
    if (wave >= total_waves) return;          // uniform per wave: EXEC stays all-1s
    int lane = threadIdx.x & 31;

    int tpb   = tiles_m * tiles_n;
    int batch = wave / tpb;
    int t     = wave - batch * tpb;
    int m0    = (t / tiles_n) * 64;
    int n0    = (t % tiles_n) * 16;

    const float* Ab = Am + (long)batch * a_batch;
    const float* Bb = Bm + (long)batch * b_batch;
    float*       Cb = Cm + (long)batch * c_batch;

    int half = lane >> 4;     // 0 -> K pair {0,1}; 1 -> K pair {2,3}
    int l    = lane & 15;     // row within A sub-tile / col of B,C tile

    const float* ar0 = Ab + (long)(m0 +  0 + l) * lda;
    const float* ar1 = Ab + (long)(m0 + 16 + l) * lda;
    const float* ar2 = Ab + (long)(m0 + 32 + l) * lda;
    const float* ar3 = Ab + (long)(m0 + 48 + l) * lda;
    long coln = (long)(n0 + l) * bs_n;

    v8f c0 = {}, c1 = {}, c2 = {}, c3 = {};
    for (int k0 = 0; k0 < Kdim; k0 += 4) {
        int ka = k0 + half * 2;
        v2f b;
        long bi = (long)ka * bs_k + coln;
        b.x = Bb[bi];
        b.y = Bb[bi + bs_k];
        v2f a0, a1, a2, a3;
        a0.x = ar0[ka]; a0.y = ar0[ka + 1];
        a1.x = ar1[ka]; a1.y = ar1[ka + 1];
        a2.x = ar2[ka]; a2.y = ar2[ka + 1];
        a3.x = ar3[ka]; a3.y = ar3[ka + 1];
        c0 = __builtin_amdgcn_wmma_f32_16x16x4_f32(false, a0, false, b, (short)0, c0, false, false);
        c1 = __builtin_amdgcn_wmma_f32_16x16x4_f32(false, a1, false, b, (short)0, c1, false, false);
        c2 = __builtin_amdgcn_wmma_f32_16x16x4_f32(false, a2, false, b, (short)0, c2, false, false);
        c3 = __builtin_amdgcn_wmma_f32_16x16x4_f32(false, a3, false, b, (short)0, c3, false, false);
    }

    float bv = bias ? bias[n0 + l] : 0.0f;
    int mb = m0 + half * 8;   // lanes 0-15: M rows 0..7; lanes 16-31: rows 8..15
#pragma unroll
    for (int r = 0; r < 8; ++r) {
        Cb[(long)(mb +  0 + r) * ldc + (n0 + l)] = c0[r] + bv;
        Cb[(long)(mb + 16 + r) * ldc + (n0 + l)] = c1[r] + bv;
        Cb[(long)(mb + 32 + r) * ldc + (n0 + l)] = c2[r] + bv;
        Cb[(long)(mb + 48 + r) * ldc + (n0 + l)] = c3[r] + bv;
    }
}

// ---------------------------------------------------------------------------
// score[b,m,n] = sum_a tanh(k1p[b,m,a] + k2p[b,n,a]) * w_s[a] + b_s
// Block handles a 16(m) x 64(n) tile; k1p/k2p staged to LDS via the CDNA5
// async global->LDS path (VGPR-free, ASYNCcnt-tracked) when available.
// LDS row stride 132 floats: 16B-aligned for B128 stores, <=2-way bank
// conflicts on the per-lane row reads in the tanh loop.
// ---------------------------------------------------------------------------
#define LDS_STRIDE (A_ + 4)

__device__ __forceinline__ void stage_tile(const float* __restrict__ gbase,
                                           float* lbase, int rows, int tid)
{
    for (int i = tid; i < rows * (A_ / 4); i += 256) {
        int r = i >> 5;           // A_/4 == 32 chunks per row
        int cch = i & 31;
        const float* g = gbase + (long)r * A_ + cch * 4;
        float* l = lbase + r * LDS_STRIDE + cch * 4;
#if __has_builtin(__builtin_amdgcn_global_load_async_to_lds_b128)
        __builtin_amdgcn_global_load_async_to_lds_b128(
            (gv4i*)g, (lv4i*)l, 0, 0);
#else
        float4 tmp = *(const float4*)g;
        *(float4*)l = tmp;
#endif
    }
}

__global__ __launch_bounds__(256)
void score_kernel(const float* __restrict__ k1p, const float* __restrict__ k2p,
                  const float* __restrict__ w_s, const float* __restrict__ b_s,
                  float* __restrict__ score)
{
    __shared__ float sk1[16 * LDS_STRIDE];
    __shared__ float sk2[64 * LDS_STRIDE];
    __shared__ float sws[A_];

    int b   = blockIdx.z;
    int m0  = blockIdx.y * 16;
    int n0  = blockIdx.x * 64;
    int tid = threadIdx.x;

    if (tid < A_) sws[tid] = w_s[tid];

    stage_tile(k1p + ((long)b * N1_ + m0) * A_, sk1, 16, tid);
    stage_tile(k2p + ((long)b * N2_ + n0) * A_, sk2, 64, tid);
    asm volatile("s_wait_asynccnt 0" ::: "memory");
    __syncthreads();

    float bs = b_s[0];
    int nl = tid & 63;        // local n
    int mg = tid >> 6;        // m group (0..3), 4 rows each

    float* srow = score + (long)b * N1_ * N2_;
#pragma unroll
    for (int j = 0; j < 4; ++j) {
        int ml = mg * 4 + j;
        const float* r1 = &sk1[ml * LDS_STRIDE];
        const float* r2 = &sk2[nl * LDS_STRIDE];
        float acc = 0.0f;
#pragma unroll 4
        for (int a = 0; a < A_; ++a)
            acc += tanhf(r1[a] + r2[a]) * sws[a];
        srow[(long)(m0 + ml) * N2_ + (n0 + nl)] = acc + bs;
    }
}

// ---------------------------------------------------------------------------
// Per-row (over N2) max and sum(exp) — for softmax(score, axis=2).
// ---------------------------------------------------------------------------
__global__ __launch_bounds__(256)
void reduce_row_kernel(const float* __restrict__ score,
                       float* __restrict__ rmax, float* __restrict__ rsum)
{
    __shared__ float red[256];
    long row = blockIdx.x;                 // b*N1 + m
    const float* s = score + row * N2_;
    int tid = threadIdx.x;

    float v0 = s[tid], v1 = s[tid + 256], v2 = s[tid + 512], v3 = s[tid + 768];
    float mx = fmaxf(fmaxf(v0, v1), fmaxf(v2, v3));
    red[tid] = mx; __syncthreads();
    for (int off = 128; off > 0; off >>= 1) {
        if (tid < off) red[tid] = fmaxf(red[tid], red[tid + off]);
        __syncthreads();
    }
    mx = red[0]; __syncthreads();

    float sum = __expf(v0 - mx) + __expf(v1 - mx) + __expf(v2 - mx) + __expf(v3 - mx);
    red[tid] = sum; __syncthreads();
    for (int off = 128; off > 0; off >>= 1) {
        if (tid < off) red[tid] += red[tid + off];
        __syncthreads();
    }
    if (tid == 0) { rmax[row] = mx; rsum[row] = red[0]; }
}

// ---------------------------------------------------------------------------
// Per-column (over N1) max and sum(exp) — for softmax(score, axis=1).
// Block (32,8): x indexes 32 consecutive columns (coalesced), y splits N1.
// ---------------------------------------------------------------------------
__global__ void reduce_col_kernel(const float* __restrict__ score,
                                  float* __restrict__ cmax, float* __restrict__ csum)
{
    __shared__ float red[8][33];
    int b = blockIdx.y;
    int n = blockIdx.x * 32 + threadIdx.x;
    int y = threadIdx.y;                       // 0..7, 32 m-values each
    const float* s = score + (long)b * N1_ * N2_ + n;

    float mx = -INFINITY;
    for (int i = 0; i < 32; ++i)
        mx = fmaxf(mx, s[(long)(y * 32 + i) * N2_]);
    red[y][threadIdx.x] = mx;
    __syncthreads();
    if (y == 0) {
        for (int k = 1; k < 8; ++k) mx = fmaxf(mx, red[k][threadIdx.x]);
        red[0][threadIdx.x] = mx;
    }
    __syncthreads();
    mx = red[0][threadIdx.x];
    __syncthreads();

    float sum = 0.0f;
    for (int i = 0; i < 32; ++i)
        sum += __expf(s[(long)(y * 32 + i) * N2_] - mx);
    red[y][threadIdx.x] = sum;
    __syncthreads();
    if (y == 0) {
        for (int k = 1; k < 8; ++k) sum += red[k][threadIdx.x];
        cmax[b * N2_ + n] = mx;
        csum[b * N2_ + n] = sum;
    }
}

// ---------------------------------------------------------------------------
// w2[b,m,n] = exp(s - cmax[b,n]) / csum[b,n]       (softmax over N1)
// w1[b,n,m] = exp(s - rmax[b,m]) / rsum[b,m]       (softmax over N2, transposed)
// ---------------------------------------------------------------------------
__global__ __launch_bounds__(256)
void normalize_kernel(const float* __restrict__ score,
                      const float* __restrict__ rmax, const float* __restrict__ rsum,
                      const float* __restrict__ cmax, const float* __restrict__ csum,
                      float* __restrict__ w1, float* __restrict__ w2)
{
    int b = blockIdx.y;
    int m = blockIdx.x;
    long row = (long)b * N1_ + m;
    const float* s = score + row * N2_;
    float rm = rmax[row];
    float rs = 1.0f / rsum[row];
    for (int n = threadIdx.x; n < N2_; n += 256) {
        float sv  = s[n];
        float w1v = __expf(sv - rm) * rs;
        float w2v = __expf(sv - cmax[b * N2_ + n]) / csum[b * N2_ + n];
        w2[row * N2_ + n] = w2v;                        // [B,N1,N2] coalesced
        w1[((long)b * N2_ + n) * N1_ + m] = w1v;        // [B,N2,N1] transposed
    }
}

// ---------------------------------------------------------------------------
extern "C" void kernel_launch(void* const* d_in, const int* in_sizes, int n_in,
                              void* d_out, int out_size, void* d_ws, size_t ws_size,
                              hipStream_t stream)
{
    const float* k1   = (const float*)d_in[0];
    const float* k2   = (const float*)d_in[1];
    const float* v1   = (const float*)d_in[2];
    const float* v2   = (const float*)d_in[3];
    const float* W_k1 = (const float*)d_in[4];
    const float* b_k1 = (const float*)d_in[5];
    const float* W_k2 = (const float*)d_in[6];
    const float* b_k2 = (const float*)d_in[7];
    const float* w_s  = (const float*)d_in[8];
    const float* b_s  = (const float*)d_in[9];

    // Output layout: o1 [B,N2,V], o2 [B,N1,V], w1 [B,N2,N1], w2 [B,N1,N2]
    float* out = (float*)d_out;
    float* o1  = out;
    float* o2  = o1 + (long)B_ * N2_ * V_;
    float* w1  = o2 + (long)B_ * N1_ * V_;
    float* w2  = w1 + (long)B_ * N2_ * N1_;

    // Workspace layout
    float* ws    = (float*)d_ws;
    float* k1p   = ws;                                 // B*N1*A
    float* k2p   = k1p + (long)B_ * N1_ * A_;          // B*N2*A
    float* score = k2p + (long)B_ * N2_ * A_;          // B*N1*N2
    float* rmax  = score + (long)B_ * N1_ * N2_;       // B*N1
    float* rsum  = rmax + B_ * N1_;                    // B*N1
    float* cmax  = rsum + B_ * N1_;                    // B*N2
    float* csum  = cmax + B_ * N2_;                    // B*N2

    // 1) k1p = k1 @ W_k1^T + b_k1   (treat [B,N1,K] as [B*N1, K])
    {
        int tm = (B_ * N1_) / 64, tn = A_ / 16;
        int waves = tm * tn;
        gemm_f32_wmma<<<(waves * 32 + 255) / 256, 256, 0, stream>>>(
            k1, W_k1, b_k1, k1p, 0, 0, 0,
            K_, /*lda=*/K_, /*bs_k=*/1, /*bs_n=*/K_, /*ldc=*/A_, tm, tn, waves);
    }
    // 2) k2p = k2 @ W_k2^T + b_k2
    {
        int tm = (B_ * N2_) / 64, tn = A_ / 16;
        int waves = tm * tn;
        gemm_f32_wmma<<<(waves * 32 + 255) / 256, 256, 0, stream>>>(
            k2, W_k2, b_k2, k2p, 0, 0, 0,
            K_, K_, 1, K_, A_, tm, tn, waves);
    }
    // 3) score
    score_kernel<<<dim3(N2_ / 64, N1_ / 16, B_), 256, 0, stream>>>(
        k1p, k2p, w_s, b_s, score);
    // 4) softmax statistics
    reduce_row_kernel<<<B_ * N1_, 256, 0, stream>>>(score, rmax, rsum);
    reduce_col_kernel<<<dim3(N2_ / 32, B_), dim3(32, 8), 0, stream>>>(score, cmax, csum);
    // 5) normalize -> w1 (transposed), w2
    normalize_kernel<<<dim3(N1_, B_), 256, 0, stream>>>(
        score, rmax, rsum, cmax, csum, w1, w2);
    // 6) o1 = w1 @ v1   ([B,N2,N1] x [B,N1,V])
    {
        int tm = N2_ / 64, tn = V_ / 16;
        int waves = B_ * tm * tn;
        gemm_f32_wmma<<<(waves * 32 + 255) / 256, 256, 0, stream>>>(
            w1, v1, nullptr, o1,
            (long)N2_ * N1_, (long)N1_ * V_, (long)N2_ * V_,
            /*Kdim=*/N1_, /*lda=*/N1_, /*bs_k=*/V_, /*bs_n=*/1, /*ldc=*/V_,
            tm, tn, waves);
    }
    // 7) o2 = w2 @ v2   ([B,N1,N2] x [B,N2,V])
    {
        int tm = N1_ / 64, tn = V_ / 16;
        int waves = B_ * tm * tn;
        gemm_f32_wmma<<<(waves * 32 + 255) / 256, 256, 0, stream>>>(
            w2, v2, nullptr, o2,
            (long)N1_ * N2_, (long)N2_ * V_, (long)N1_ * V_,
            /*Kdim=*/N2_, /*lda=*/N2_, /*bs_k=*/V_, /*bs_n=*/1, /*ldc=*/V_,
            tm, tn, waves);
    }
}